// SocialLSTMClassifier_38560216384112
// MI455X (gfx1250) — compile-verified
//
#include <hip/hip_runtime.h>
#include <math.h>

// ---- problem constants (from reference) ----
constexpr int   kH  = 128;     // hidden
constexpr int   kD  = 8;       // input feature
constexpr int   kT  = 15;      // obs_len
constexpr int   kN  = 16384;   // neighbors
constexpr int   kGX = 4, kGY = 4;
constexpr int   kC  = 16;      // grid cells
constexpr float kScale  = 0.08838834764831845f;  // 1/sqrt(128)
constexpr float kHalfNS = 2.0f;                  // NS/2
constexpr float kCW = 1.0f, kCH = 1.0f;          // NS/GX, NS/GY

// ---- workspace layout (floats) ----
constexpr int WS_H14  = 0;     // [128] h at t=T-1
constexpr int WS_A    = 128;   // [128] a = scale * Wk^T q
constexpr int WS_QG   = 256;   // [128] grid query
constexpr int WS_BSC  = 384;   // [1]   scale * (q . bk)
constexpr int WS_PART = 512;   // per-block per-cell partials
constexpr int PART_STRIDE = 132;  // {max, sumexp, count, acc[128], pad}

constexpr int K2_WAVES = 8;
constexpr int K2_NBPB  = K2_WAVES * 16;   // 128 neighbors / block
constexpr int K2_NBLK  = kN / K2_NBPB;    // 128 blocks

typedef __attribute__((ext_vector_type(2))) float v2f;
typedef __attribute__((ext_vector_type(8))) float v8f;

constexpr float kLog2e = 1.4426950408889634f;

// ---- fast transcendentals: map to single CDNA5 TRANS ops where available ----
__device__ __forceinline__ float fast_tanh(float x) {
#if __has_builtin(__builtin_amdgcn_tanhf)
    return __builtin_amdgcn_tanhf(x);          // v_tanh_f32
#else
    return tanhf(x);
#endif
}

__device__ __forceinline__ float fast_rcp(float x) {
#if __has_builtin(__builtin_amdgcn_rcpf)
    return __builtin_amdgcn_rcpf(x);           // v_rcp_f32
#else
    return 1.0f / x;
#endif
}

__device__ __forceinline__ float fast_exp(float x) {
#if __has_builtin(__builtin_amdgcn_exp2f)
    return __builtin_amdgcn_exp2f(x * kLog2e); // v_exp_f32
#else
    return __expf(x);
#endif
}

__device__ __forceinline__ float fast_sigmoid(float x) {
    return fast_rcp(1.0f + fast_exp(-x));
}

// =====================================================================
// Kernel 1: target LSTM chain (sequential, tiny) + query precomputation
// =====================================================================
__global__ void __launch_bounds__(128)
k1_target(const float* __restrict__ target,
          const float* __restrict__ W_ih, const float* __restrict__ W_hh,
          const float* __restrict__ b_ih, const float* __restrict__ b_hh,
          const float* __restrict__ Wq,  const float* __restrict__ bq,
          const float* __restrict__ Wk,  const float* __restrict__ bk,
          const float* __restrict__ Wgq, const float* __restrict__ bgq,
          float* __restrict__ ws)
{
    __shared__ float h_sh[kH], c_sh[kH], q_sh[kH];
    const int j = threadIdx.x;
    h_sh[j] = 0.0f; c_sh[j] = 0.0f;
    __syncthreads();

    for (int t = 0; t < kT; ++t) {
        float x[kD];
#pragma unroll
        for (int d = 0; d < kD; ++d) x[d] = target[t * kD + d];
        float g[4];
#pragma unroll
        for (int r = 0; r < 4; ++r) {
            const int k = r * kH + j;
            float acc = b_ih[k] + b_hh[k];
#pragma unroll
            for (int d = 0; d < kD; ++d) acc += x[d] * W_ih[k * kD + d];
            for (int i = 0; i < kH; ++i) acc += h_sh[i] * W_hh[k * kH + i];
            g[r] = acc;
        }
        __syncthreads();  // all reads of h_sh done
        const float c2 = fast_sigmoid(g[1]) * c_sh[j] + fast_sigmoid(g[0]) * fast_tanh(g[2]);
        const float h2 = fast_sigmoid(g[3]) * fast_tanh(c2);
        c_sh[j] = c2; h_sh[j] = h2;
        __syncthreads();
    }

    // q = Wq h + bq
    float q = bq[j];
    for (int i = 0; i < kH; ++i) q += Wq[j * kH + i] * h_sh[i];
    q_sh[j] = q;
    __syncthreads();

    // a = scale * Wk^T q  (column dot)
    float a = 0.0f;
    for (int i = 0; i < kH; ++i) a += Wk[i * kH + j] * q_sh[i];
    a *= kScale;

    // qg = Wgq h + bgq
    float qg = bgq[j];
    for (int i = 0; i < kH; ++i) qg += Wgq[j * kH + i] * h_sh[i];

    ws[WS_H14 + j] = h_sh[j];
    ws[WS_A   + j] = a;
    ws[WS_QG  + j] = qg;
    if (j == 0) {
        float bs = 0.0f;
        for (int i = 0; i < kH; ++i) bs += q_sh[i] * bk[i];
        ws[WS_BSC] = bs * kScale;
    }
}

// =====================================================================
// Kernel 2: fused neighbor LSTM (WMMA f32 16x16x4) + scores +
//           per-cell flash-softmax partials (per block)
// =====================================================================
__global__ void __launch_bounds__(256)
k2_neighbors(const float* __restrict__ target,
             const float* __restrict__ others,
             const int*   __restrict__ nmask,
             const float* __restrict__ W_ih,
             const float* __restrict__ b_ih, const float* __restrict__ b_hh,
             float* __restrict__ ws)
{
    __shared__ float s_sh[K2_NBPB];
    __shared__ int   cell_sh[K2_NBPB];
    __shared__ float e_sh[K2_NBPB];
    __shared__ float mx_sh[kC];
    __shared__ float S_sh[kC];
    __shared__ int   cnt_sh[kC];
    __shared__ float acc_sh[kC * kH];

    const int tid   = threadIdx.x;
    const int wave  = tid >> 5;
    const int lane  = tid & 31;
    const int lo    = lane & 15;   // N-column / A-row selector
    const int hi    = lane >> 4;   // K-half selector
    const int wtile = wave * 16;
    const int nbase = blockIdx.x * K2_NBPB;

    for (int q = tid; q < kC * kH; q += blockDim.x) acc_sh[q] = 0.0f;
    if (tid < kC) S_sh[tid] = 0.0f;

    const float  bscal = ws[WS_BSC];
    const size_t obase = (size_t)(kT - 1) * kN * kD;

    // ---- A operand: 16 neighbors x K=8 (two 16x4 fp32 tiles) ----
    // layout: lane holds A[lo][2*hi + {0,1}] (tile0: K 0..3, tile1: K 4..7)
    const size_t arow = obase + (size_t)(nbase + wtile + lo) * kD;
    v2f A0, A1;
    A0[0] = others[arow + 2 * hi + 0];
    A0[1] = others[arow + 2 * hi + 1];
    A1[0] = others[arow + 4 + 2 * hi + 0];
    A1[1] = others[arow + 4 + 2 * hi + 1];

    float hoReg[8][8];   // ho tile values per column-group
    float sacc[8];
#pragma unroll
    for (int v = 0; v < 8; ++v) sacc[v] = 0.0f;

#pragma unroll
    for (int cg = 0; cg < 8; ++cg) {
        const int j0 = cg * 16;
        const int jn = j0 + lo;          // this lane's gate column within H
        v8f acc_i = {0,0,0,0,0,0,0,0};
        v8f acc_g = {0,0,0,0,0,0,0,0};
        v8f acc_o = {0,0,0,0,0,0,0,0};
        // B operand: B[k][n] = W_ih[col + n][k]; lane holds W_ih[jn][2*hi + {0,1}]
        {
            const int col = jn;                         // i-gates (rows 0..127)
            v2f B0, B1;
            B0[0] = W_ih[col * kD + 2 * hi + 0]; B0[1] = W_ih[col * kD + 2 * hi + 1];
            B1[0] = W_ih[col * kD + 4 + 2 * hi + 0]; B1[1] = W_ih[col * kD + 4 + 2 * hi + 1];
            acc_i = __builtin_amdgcn_wmma_f32_16x16x4_f32(false, A0, false, B0, (short)0, acc_i, false, false);
            acc_i = __builtin_amdgcn_wmma_f32_16x16x4_f32(false, A1, false, B1, (short)0, acc_i, false, false);
        }
        {
            const int col = 256 + jn;                   // g-gates
            v2f B0, B1;
            B0[0] = W_ih[col * kD + 2 * hi + 0]; B0[1] = W_ih[col * kD + 2 * hi + 1];
            B1[0] = W_ih[col * kD + 4 + 2 * hi + 0]; B1[1] = W_ih[col * kD + 4 + 2 * hi + 1];
            acc_g = __builtin_amdgcn_wmma_f32_16x16x4_f32(false, A0, false, B0, (short)0, acc_g, false, false);
            acc_g = __builtin_amdgcn_wmma_f32_16x16x4_f32(false, A1, false, B1, (short)0, acc_g, false, false);
        }
        {
            const int col = 384 + jn;                   // o-gates (f-gate dead: c0 == 0)
            v2f B0, B1;
            B0[0] = W_ih[col * kD + 2 * hi + 0]; B0[1] = W_ih[col * kD + 2 * hi + 1];
            B1[0] = W_ih[col * kD + 4 + 2 * hi + 0]; B1[1] = W_ih[col * kD + 4 + 2 * hi + 1];
            acc_o = __builtin_amdgcn_wmma_f32_16x16x4_f32(false, A0, false, B0, (short)0, acc_o, false, false);
            acc_o = __builtin_amdgcn_wmma_f32_16x16x4_f32(false, A1, false, B1, (short)0, acc_o, false, false);
        }
        const float bi  = b_ih[jn]       + b_hh[jn];
        const float bg  = b_ih[jn + 256] + b_hh[jn + 256];
        const float bo_ = b_ih[jn + 384] + b_hh[jn + 384];
        const float aj  = ws[WS_A + jn];
        // D layout: lane(lo,hi), VGPR v -> element [m = v + 8*hi][n = lo]
#pragma unroll
        for (int v = 0; v < 8; ++v) {
            const float gi = acc_i[v] + bi;
            const float gg = acc_g[v] + bg;
            const float go = acc_o[v] + bo_;
            const float cc = fast_sigmoid(gi) * fast_tanh(gg);
            const float ho = fast_sigmoid(go) * fast_tanh(cc);
            hoReg[cg][v] = ho;
            sacc[v] += aj * ho;                 // partial of a . ho
        }
    }

    // reduce score partials across the 16 lanes of each half-wave
#pragma unroll
    for (int v = 0; v < 8; ++v) {
        float s = sacc[v];
        s += __shfl_xor(s, 1, 16);
        s += __shfl_xor(s, 2, 16);
        s += __shfl_xor(s, 4, 16);
        s += __shfl_xor(s, 8, 16);
        sacc[v] = s + bscal;                    // + scale*(q.bk)
    }
    if (lo == 0) {
#pragma unroll
        for (int v = 0; v < 8; ++v) s_sh[wtile + hi * 8 + v] = sacc[v];
    }
    // lanes 0..15: grid binning for neighbor m = lo
    if (hi == 0) {
        const int g = nbase + wtile + lo;
        const float px = others[obase + (size_t)g * kD + 0];
        const float py = others[obase + (size_t)g * kD + 1];
        const float tx = target[(kT - 1) * kD + 0];
        const float ty = target[(kT - 1) * kD + 1];
        const float rx = px - tx, ry = py - ty;
        const int cx = (int)(rx / kCW) + kGX / 2;   // trunc toward zero
        const int cy = (int)(ry / kCH) + kGY / 2;
        const bool ok = (fabsf(rx) <= kHalfNS) && (fabsf(ry) <= kHalfNS) &&
                        (cx >= 0) && (cx < kGX) && (cy >= 0) && (cy < kGY) &&
                        (nmask[(kT - 1) * kN + g] != 0);
        cell_sh[wtile + lo] = ok ? (cy * kGX + cx) : -1;
    }
    __syncthreads();

    // per-cell block max + member count (cell owners)
    if (tid < kC) {
        float mxv = -3.0e38f; int cnt = 0;
        for (int n = 0; n < K2_NBPB; ++n)
            if (cell_sh[n] == tid) { mxv = fmaxf(mxv, s_sh[n]); ++cnt; }
        mx_sh[tid] = mxv; cnt_sh[tid] = cnt;
    }
    __syncthreads();

    // e = exp(s - max), sumexp
    if (tid < K2_NBPB) {
        const int cc = cell_sh[tid];
        float e = 0.0f;
        if (cc >= 0) {
            e = fast_exp(s_sh[tid] - mx_sh[cc]);
            atomicAdd(&S_sh[cc], e);
        }
        e_sh[tid] = e;
    }
    __syncthreads();

    // acc[cell][j] += e * ho[m][j]
#pragma unroll
    for (int cg = 0; cg < 8; ++cg) {
#pragma unroll
        for (int v = 0; v < 8; ++v) {
            const int mm = wtile + hi * 8 + v;
            const int cc = cell_sh[mm];
            if (cc >= 0)
                atomicAdd(&acc_sh[cc * kH + cg * 16 + lo], e_sh[mm] * hoReg[cg][v]);
        }
    }
    __syncthreads();

    // write per-block partials
    float* part = ws + WS_PART + (size_t)blockIdx.x * kC * PART_STRIDE;
    if (tid < kC) {
        part[tid * PART_STRIDE + 0] = mx_sh[tid];
        part[tid * PART_STRIDE + 1] = S_sh[tid];
        part[tid * PART_STRIDE + 2] = (float)cnt_sh[tid];
    }
    for (int q = tid; q < kC * kH; q += blockDim.x) {
        const int c = q >> 7, jj = q & 127;
        part[c * PART_STRIDE + 3 + jj] = acc_sh[q];
    }
}

// =====================================================================
// Kernel 3: merge partials (LSE), cell outputs, grid attention, classify
// =====================================================================
__global__ void __launch_bounds__(128)
k3_finalize(const float* __restrict__ Wv,  const float* __restrict__ bv,
            const float* __restrict__ Wo,  const float* __restrict__ bo,
            const float* __restrict__ Wgk, const float* __restrict__ bgk,
            const float* __restrict__ Wgv, const float* __restrict__ bgv,
            const float* __restrict__ Wc,  const float* __restrict__ bc,
            const float* __restrict__ ws, int nblocks,
            float* __restrict__ out)
{
    __shared__ float Mc[kC], Sc[kC], Wsum[kC];
    __shared__ int   Cnt[kC];
    __shared__ float bufA[kC * kH], bufB[kC * kH];
    __shared__ float sc_sh[kC];
    __shared__ float comb[kH];
    const int j = threadIdx.x;
    const float* part = ws + WS_PART;

    if (j < kC) {
        float mxv = -3.0e38f; int cnt = 0;
        for (int b = 0; b < nblocks; ++b) {
            const float* p = part + ((size_t)b * kC + j) * PART_STRIDE;
            const int c = (int)p[2];
            if (c > 0) { mxv = fmaxf(mxv, p[0]); cnt += c; }
        }
        Mc[j] = mxv; Cnt[j] = cnt;
        float s = 0.0f;
        for (int b = 0; b < nblocks; ++b) {
            const float* p = part + ((size_t)b * kC + j) * PART_STRIDE;
            if ((int)p[2] > 0) s += p[1] * fast_exp(p[0] - mxv);
        }
        Sc[j] = s;
        Wsum[j] = s / fmaxf(s, 1e-30f);   // == sum(w) per cell
    }
    __syncthreads();

    // u[c] = (sum e*ho)/max(S,1e-30)  -> bufA
    for (int c = 0; c < kC; ++c) {
        float a = 0.0f;
        if (Cnt[c] > 0) {
            for (int b = 0; b < nblocks; ++b) {
                const float* p = part + ((size_t)b * kC + c) * PART_STRIDE;
                if ((int)p[2] > 0) a += p[3 + j] * fast_exp(p[0] - Mc[c]);
            }
            a /= fmaxf(Sc[c], 1e-30f);
        }
        bufA[c * kH + j] = a;
    }
    __syncthreads();

    // t = Wv u + sum(w)*bv -> bufB
    for (int c = 0; c < kC; ++c) {
        float acc = Wsum[c] * bv[j];
        for (int i = 0; i < kH; ++i) acc += Wv[j * kH + i] * bufA[c * kH + i];
        bufB[c * kH + j] = acc;
    }
    __syncthreads();

    // cell_out = nonempty ? Wo t + bo : 0 -> bufA
    for (int c = 0; c < kC; ++c) {
        float acc = 0.0f;
        if (Cnt[c] > 0) {
            acc = bo[j];
            for (int i = 0; i < kH; ++i) acc += Wo[j * kH + i] * bufB[c * kH + i];
        }
        bufA[c * kH + j] = acc;
    }
    __syncthreads();

    // K = Wgk cell_out + bgk -> bufB
    for (int c = 0; c < kC; ++c) {
        float acc = bgk[j];
        for (int i = 0; i < kH; ++i) acc += Wgk[j * kH + i] * bufA[c * kH + i];
        bufB[c * kH + j] = acc;
    }
    __syncthreads();
    if (j < kC) {
        float acc = 0.0f;
        for (int i = 0; i < kH; ++i) acc += ws[WS_QG + i] * bufB[j * kH + i];
        sc_sh[j] = acc * kScale;
    }
    __syncthreads();

    // softmax over 16 cells (every thread replicates)
    float gmax = -3.0e38f;
    for (int c = 0; c < kC; ++c) gmax = fmaxf(gmax, sc_sh[c]);
    float gw[kC]; float gsum = 0.0f;
    for (int c = 0; c < kC; ++c) { gw[c] = fast_exp(sc_sh[c] - gmax); gsum += gw[c]; }
    const float inv = 1.0f / gsum;
    __syncthreads();

    // V = Wgv cell_out + bgv -> bufB (overwrite; reads bufA)
    for (int c = 0; c < kC; ++c) {
        float acc = bgv[j];
        for (int i = 0; i < kH; ++i) acc += Wgv[j * kH + i] * bufA[c * kH + i];
        bufB[c * kH + j] = acc;
    }
    __syncthreads();

    float ctx = 0.0f;
    for (int c = 0; c < kC; ++c) ctx += gw[c] * inv * bufB[c * kH + j];
    comb[j] = ws[WS_H14 + j] + ctx;
    __syncthreads();

    if (j < 2) {
        float acc = bc[j];
        for (int i = 0; i < kH; ++i) acc += Wc[j * kH + i] * comb[i];
        out[j] = acc;
    }
}

// =====================================================================
extern "C" void kernel_launch(void* const* d_in, const int* in_sizes, int n_in,
                              void* d_out, int out_size, void* d_ws, size_t ws_size,
                              hipStream_t stream)
{
    const float* target = (const float*)d_in[0];
    const float* others = (const float*)d_in[1];
    const int*   nmask  = (const int*)  d_in[2];
    const float* W_ih = (const float*)d_in[3];
    const float* W_hh = (const float*)d_in[4];
    const float* b_ih = (const float*)d_in[5];
    const float* b_hh = (const float*)d_in[6];
    const float* Wq  = (const float*)d_in[7];   const float* bq  = (const float*)d_in[8];
    const float* Wk  = (const float*)d_in[9];   const float* bk  = (const float*)d_in[10];
    const float* Wv  = (const float*)d_in[11];  const float* bv  = (const float*)d_in[12];
    const float* Wo  = (const float*)d_in[13];  const float* bo  = (const float*)d_in[14];
    const float* Wgq = (const float*)d_in[15];  const float* bgq = (const float*)d_in[16];
    const float* Wgk = (const float*)d_in[17];  const float* bgk = (const float*)d_in[18];
    const float* Wgv = (const float*)d_in[19];  const float* bgv = (const float*)d_in[20];
    const float* Wc  = (const float*)d_in[21];  const float* bc  = (const float*)d_in[22];
    float* ws  = (float*)d_ws;
    float* out = (float*)d_out;

    hipLaunchKernelGGL(k1_target, dim3(1), dim3(128), 0, stream,
                       target, W_ih, W_hh, b_ih, b_hh, Wq, bq, Wk, bk, Wgq, bgq, ws);
    hipLaunchKernelGGL(k2_neighbors, dim3(K2_NBLK), dim3(256), 0, stream,
                       target, others, nmask, W_ih, b_ih, b_hh, ws);
    hipLaunchKernelGGL(k3_finalize, dim3(1), dim3(128), 0, stream,
                       Wv, bv, Wo, bo, Wgk, bgk, Wgv, bgv, Wc, bc, ws, K2_NBLK, out);
}